// PointGroup_2508260901477
// MI455X (gfx1250) — compile-verified
//
#include <hip/hip_runtime.h>
#include <hip/hip_bf16.h>
#include <stdint.h>

// Problem constants (match reference)
#define NPTS   200000
#define NCLS   20
#define PNUM   256
#define MEDGE  400000
#define THRESH 100          // CLUSTER_PROPOSE_POINTS
#define NW     6250         // NPTS/32 mask words per proposal (exact)
#define KSLICES 64
#define WPW     98          // ceil(NW/KSLICES)
#define SCOLS   32          // cols in S: 0..19 classes, 20 = ones(count), 21..31 zero
#define PPAD    32          // padded probs row stride (floats)

typedef float v2f __attribute__((ext_vector_type(2)));
typedef float v4f __attribute__((ext_vector_type(4)));
typedef float v8f __attribute__((ext_vector_type(8)));

// ---------------- init: zero maskbits + Spart + S, rep = INT_MAX ----------------
#define ZWORDS 2132480   // (6,400,000 + 2,097,152 + 32,768) / 4
__global__ void k_init(unsigned* __restrict__ zbase, int* __restrict__ rep) {
    int t = blockIdx.x * blockDim.x + threadIdx.x;
    if (t < ZWORDS) zbase[t] = 0u;
    if (t < PNUM)   rep[t] = 0x7FFFFFFF;
}

// ---------------- per point: softmax, argmax, bias copy, padded probs ----------------
__global__ void k_point(const float* __restrict__ logit, const float* __restrict__ bias,
                        float* __restrict__ out_probs, float* __restrict__ out_bias,
                        float* __restrict__ probs_pad, int* __restrict__ seg) {
    int n = blockIdx.x * blockDim.x + threadIdx.x;
    if (n >= NPTS) return;
    const v4f* l4 = (const v4f*)(logit + (size_t)n * NCLS);
    float v[NCLS];
#pragma unroll
    for (int i = 0; i < 5; ++i) {
        v4f x = l4[i];
        v[4*i+0] = x.x; v[4*i+1] = x.y; v[4*i+2] = x.z; v[4*i+3] = x.w;
    }
    float mx = v[0]; int bi = 0;
#pragma unroll
    for (int i = 1; i < NCLS; ++i) if (v[i] > mx) { mx = v[i]; bi = i; }  // first max, like argmax
    float s = 0.f;
#pragma unroll
    for (int i = 0; i < NCLS; ++i) { v[i] = __expf(v[i] - mx); s += v[i]; }
    float inv = 1.0f / s;
#pragma unroll
    for (int i = 0; i < NCLS; ++i) v[i] *= inv;

    // final probs output (20 cols)
    v4f* o4 = (v4f*)(out_probs + (size_t)n * NCLS);
#pragma unroll
    for (int i = 0; i < 5; ++i) {
        v4f x; x.x = v[4*i+0]; x.y = v[4*i+1]; x.z = v[4*i+2]; x.w = v[4*i+3];
        o4[i] = x;
    }
    // padded probs for WMMA B-matrix: cols 0..19 probs, col 20 = 1.0 (count col), 21..31 = 0
    v4f* p4 = (v4f*)(probs_pad + (size_t)n * PPAD);
#pragma unroll
    for (int i = 0; i < 5; ++i) {
        v4f x; x.x = v[4*i+0]; x.y = v[4*i+1]; x.z = v[4*i+2]; x.w = v[4*i+3];
        p4[i] = x;
    }
    v4f pad0; pad0.x = 1.0f; pad0.y = 0.f; pad0.z = 0.f; pad0.w = 0.f;
    v4f padz; padz.x = 0.f;  padz.y = 0.f; padz.z = 0.f; padz.w = 0.f;
    p4[5] = pad0; p4[6] = padz; p4[7] = padz;

    seg[n] = bi;
    out_bias[(size_t)n*3+0] = bias[(size_t)n*3+0];
    out_bias[(size_t)n*3+1] = bias[(size_t)n*3+1];
    out_bias[(size_t)n*3+2] = bias[(size_t)n*3+2];
}

// ---------------- per edge: scatter bit (dedup), min representative ----------------
__global__ void k_edge(const int* __restrict__ pid, const int* __restrict__ nid,
                       unsigned* __restrict__ maskbits, int* __restrict__ rep) {
    int m = blockIdx.x * blockDim.x + threadIdx.x;
    if (m >= MEDGE) return;
    int p = pid[m], n = nid[m];
    atomicOr(&maskbits[(size_t)p * NW + (n >> 5)], 1u << (n & 31));
    atomicMin(&rep[p], n);
}

// ---------------- WMMA: Spart[slice] = Mask(16xK) x ProbsPad(Kx32) ----------------
// one wave per (row_tile, k_slice); f32 16x16x4 WMMA, exact arithmetic
__global__ void __launch_bounds__(32) k_score_wmma(const unsigned* __restrict__ maskbits,
                                                   const float* __restrict__ probs_pad,
                                                   float* __restrict__ Spart) {
    const int lane = threadIdx.x;
    const int rt   = blockIdx.x;        // 0..15 row tile (16 proposals)
    const int sl   = blockIdx.y;        // 0..63 K slice
    const int r0   = rt * 16;
    const int col  = lane & 15;         // A row / B,D col for this lane
    const int hi   = lane >> 4;
    const int ksub = hi * 2;            // lanes 0-15 -> K {0,1}, lanes 16-31 -> K {2,3}
    v8f acc0 = {}; v8f acc1 = {};
    const int wbeg = sl * WPW;
    const int wend = min(wbeg + WPW, NW);
    const unsigned* mrow = maskbits + (size_t)(r0 + col) * NW;
    for (int w = wbeg; w < wend; ++w) {
        unsigned word = mrow[w];
        __builtin_prefetch(mrow + w + 16, 0, 1);                 // stream mask words ahead
        if (!__any(word != 0u)) continue;                        // uniform skip
        // one base pointer per word; per-step deltas fold into load immediates
        const float* pb = probs_pad + ((((size_t)w << 5) + (size_t)ksub) << 5) + col;
#pragma unroll
        for (int j = 0; j < 8; ++j) {
            unsigned nib = (word >> (j * 4)) & 0xFu;             // 4 K-bits for this row
            if (!__any((int)nib)) continue;                      // uniform skip (~60% at this density)
            v2f a; a.x = (float)((nib >> ksub) & 1u);
                   a.y = (float)((nib >> (ksub + 1)) & 1u);
            v2f b0; b0.x = pb[j * 128 + 0];  b0.y = pb[j * 128 + 32];   // classes 0..15
            v2f b1; b1.x = pb[j * 128 + 16]; b1.y = pb[j * 128 + 48];   // 16..19 | ones | zeros
            acc0 = __builtin_amdgcn_wmma_f32_16x16x4_f32(false, a, false, b0, (short)0, acc0, false, false);
            acc1 = __builtin_amdgcn_wmma_f32_16x16x4_f32(false, a, false, b1, (short)0, acc1, false, false);
        }
    }
    // D layout: lane l, VGPR v -> row = v + 8*(l>>4), col = l&15
    float* out = Spart + (size_t)sl * (PNUM * SCOLS);
#pragma unroll
    for (int v = 0; v < 8; ++v) {
        int orow = r0 + v + 8 * hi;
        out[orow * SCOLS + col]      = acc0[v];
        out[orow * SCOLS + 16 + col] = acc1[v];
    }
}

// ---------------- fixed-order slice reduction (deterministic) ----------------
__global__ void k_reduce(const float* __restrict__ Spart, float* __restrict__ S) {
    int t = blockIdx.x * blockDim.x + threadIdx.x;
    if (t >= PNUM * SCOLS) return;
    float s = 0.f;
    for (int sl = 0; sl < KSLICES; ++sl) s += Spart[(size_t)sl * (PNUM * SCOLS) + t];
    S[t] = s;
}

// ---------------- per proposal finalize ----------------
__global__ void k_finalize(const float* __restrict__ S, const int* __restrict__ rep,
                           const int* __restrict__ seg, float* __restrict__ out_scores,
                           float* __restrict__ out_classes, int* __restrict__ pmask) {
    int p = threadIdx.x;                       // 1 block of 256
    float cf  = S[p * SCOLS + NCLS];           // exact integer-valued count
    int   cnt = (int)(cf + 0.5f);
    int   rv  = rep[p];
    int   rc  = rv < 0 ? 0 : (rv > NPTS - 1 ? NPTS - 1 : rv);   // clip (INT_MAX for empty)
    int   inst = seg[rc];
    bool  ok   = cnt > THRESH;
    float score = S[p * SCOLS + inst];
    out_scores[p]  = ok ? score / cf : 0.0f;
    out_classes[p] = ok ? (float)inst : -1.0f;
    pmask[p] = ok ? 1 : 0;
}

// ---------------- big mask write: bit word -> 32 floats, streaming stores ----------------
__global__ void k_maskwrite(const unsigned* __restrict__ maskbits, const int* __restrict__ pmask,
                            float* __restrict__ out_masks) {
    int t = blockIdx.x * blockDim.x + threadIdx.x;
    if (t >= PNUM * NW) return;
    int p = t / NW;
    int w = t - p * NW;
    unsigned word = pmask[p] ? maskbits[t] : 0u;
    v4f* base = (v4f*)(out_masks + (size_t)p * NPTS + (size_t)w * 32);
#pragma unroll
    for (int q = 0; q < 8; ++q) {
        v4f f;
        f.x = (float)((word >> (q * 4 + 0)) & 1u);
        f.y = (float)((word >> (q * 4 + 1)) & 1u);
        f.z = (float)((word >> (q * 4 + 2)) & 1u);
        f.w = (float)((word >> (q * 4 + 3)) & 1u);
        __builtin_nontemporal_store(f, base + q);   // 204.8 MB streamed, bypass cache pollution
    }
}

extern "C" void kernel_launch(void* const* d_in, const int* in_sizes, int n_in,
                              void* d_out, int out_size, void* d_ws, size_t ws_size,
                              hipStream_t stream) {
    const float* logit = (const float*)d_in[0];
    const float* bias  = (const float*)d_in[1];
    // d_in[2] = coord: unused (center_pred is dead code in the reference)
    const int* pid = (const int*)d_in[3];
    const int* nid = (const int*)d_in[4];

    float* out        = (float*)d_out;
    float* out_scores = out;                                        // P
    float* out_masks  = out + PNUM;                                 // P*N
    float* out_class  = out + PNUM + (size_t)PNUM * NPTS;           // P
    float* out_bias   = out_class + PNUM;                           // N*3
    float* out_probs  = out_bias + (size_t)NPTS * 3;                // N*C

    // padded probs (N x 32 = 25.6 MB) lives in the TAIL of the pred_masks output
    // region: consumed by k_score_wmma, then fully overwritten by k_maskwrite
    // (stream-ordered), so every output element still gets its final value.
    float* probs_pad = out_masks + ((size_t)PNUM * NPTS - (size_t)NPTS * PPAD);

    // workspace layout (bytes), total ~9.4 MB
    char* ws = (char*)d_ws;
    unsigned* maskbits = (unsigned*)ws;                 // 6,400,000 B  (P*NW u32)
    float*    Spart    = (float*)(ws + 6400000);        // 2,097,152 B  (64*P*32 f32)
    float*    S        = (float*)(ws + 8497152);        //    32,768 B  (P*32 f32)
    int*      rep      = (int*)  (ws + 8529920);        //     1,024 B
    int*      pmask    = (int*)  (ws + 8530944);        //     1,024 B
    int*      seg      = (int*)  (ws + 8531968);        //   800,000 B

    k_init     <<<ZWORDS / 256, 256, 0, stream>>>(maskbits, rep);
    k_point    <<<(NPTS + 255) / 256, 256, 0, stream>>>(logit, bias, out_probs, out_bias, probs_pad, seg);
    k_edge     <<<(MEDGE + 255) / 256, 256, 0, stream>>>(pid, nid, maskbits, rep);
    k_score_wmma<<<dim3(16, KSLICES), 32, 0, stream>>>(maskbits, probs_pad, Spart);
    k_reduce   <<<(PNUM * SCOLS + 255) / 256, 256, 0, stream>>>(Spart, S);
    k_finalize <<<1, 256, 0, stream>>>(S, rep, seg, out_scores, out_class, pmask);
    k_maskwrite<<<(PNUM * NW + 255) / 256, 256, 0, stream>>>(maskbits, pmask, out_masks);
}